// SimpleGAT_34359738368168
// MI455X (gfx1250) — compile-verified
//
#include <hip/hip_runtime.h>
#include <hip/hip_bf16.h>
#include <math.h>

typedef float v2f __attribute__((ext_vector_type(2)));
typedef float v8f __attribute__((ext_vector_type(8)));

#define NEG_SLOPE 0.2f
#define NUM_GRAPHS 512
#define HDIM 32

// ---------------------------------------------------------------------------
// Per-layer scratch init: num=0, denom=0, mmax=-inf, pool buffers=0
// ---------------------------------------------------------------------------
__global__ void init_kernel(float* __restrict__ num, float* __restrict__ denom,
                            float* __restrict__ mmax, float* __restrict__ gsum,
                            float* __restrict__ gcnt, int N) {
    int i = blockIdx.x * blockDim.x + threadIdx.x;
    int total = N * HDIM;
    if (i < total) num[i] = 0.f;
    if (i < N) { denom[i] = 0.f; mmax[i] = -INFINITY; }
    if (i < NUM_GRAPHS * HDIM) gsum[i] = 0.f;
    if (i < NUM_GRAPHS) gcnt[i] = 0.f;
}

// ---------------------------------------------------------------------------
// xl = x @ Wl, xr = x @ Wr  via V_WMMA_F32_16X16X4_F32.
// Operands are staged zero-padded into LDS once so the WMMA inner loop is
// branch-free (ds_load + v_wmma only). One wave = one 16-row x 32-col tile
// for BOTH weight matrices (4 accumulators, A-fragment reuse).
// ---------------------------------------------------------------------------
template <int CIN>
__global__ void gemm_xlxr_wmma(const float* __restrict__ x,
                               const float* __restrict__ Wl,
                               const float* __restrict__ Wr,
                               float* __restrict__ xl, float* __restrict__ xr,
                               int N) {
    constexpr int KPAD = (CIN + 3) & ~3;      // 19 -> 20, 32 -> 32
    constexpr int APITCH = KPAD + 2;          // keep 8B alignment, dodge bank conflicts
    __shared__ float sW[2][KPAD][HDIM];       // zero-padded weights
    __shared__ float sA[8][16][APITCH];       // per-wave A tiles

    const int tid  = threadIdx.x;
    const int wid  = tid >> 5;
    const int lane = tid & 31;
    const int row0 = (blockIdx.x * 8 + wid) * 16;

    // Stage both weight matrices (whole block cooperates; pads zeroed).
    for (int idx = tid; idx < 2 * KPAD * HDIM; idx += 256) {
        const int m   = idx / (KPAD * HDIM);
        const int rem = idx % (KPAD * HDIM);
        const int k   = rem / HDIM;
        const int n   = rem % HDIM;
        const float* W = m ? Wr : Wl;
        sW[m][k][n] = (k < CIN) ? W[(size_t)k * HDIM + n] : 0.f;
    }

    // Stage this wave's 16 x KPAD A tile (pads zeroed).
    if (row0 < N) {
        for (int idx = lane; idx < 16 * KPAD; idx += 32) {
            const int r = idx / KPAD, c = idx % KPAD;
            const int row = row0 + r;
            sA[wid][r][c] = (row < N && c < CIN) ? x[(size_t)row * CIN + c] : 0.f;
        }
    }
    __syncthreads();
    if (row0 >= N) return;                    // wave-uniform: EXEC stays full

    const int half = lane >> 4;               // A: K+0/1 vs K+2/3 halves
    const int l16  = lane & 15;

    v8f cl0 = {}, cl1 = {}, cr0 = {}, cr1 = {};
#pragma unroll
    for (int kc = 0; kc < KPAD; kc += 4) {
        const int ka = kc + half * 2;
        v2f a;
        a.x = sA[wid][l16][ka];
        a.y = sA[wid][l16][ka + 1];
        v2f bl0, bl1, br0, br1;
        bl0.x = sW[0][ka][l16];        bl0.y = sW[0][ka + 1][l16];
        bl1.x = sW[0][ka][16 + l16];   bl1.y = sW[0][ka + 1][16 + l16];
        br0.x = sW[1][ka][l16];        br0.y = sW[1][ka + 1][l16];
        br1.x = sW[1][ka][16 + l16];   br1.y = sW[1][ka + 1][16 + l16];

        cl0 = __builtin_amdgcn_wmma_f32_16x16x4_f32(false, a, false, bl0, (short)0, cl0, false, false);
        cl1 = __builtin_amdgcn_wmma_f32_16x16x4_f32(false, a, false, bl1, (short)0, cl1, false, false);
        cr0 = __builtin_amdgcn_wmma_f32_16x16x4_f32(false, a, false, br0, (short)0, cr0, false, false);
        cr1 = __builtin_amdgcn_wmma_f32_16x16x4_f32(false, a, false, br1, (short)0, cr1, false, false);
    }

    // C/D layout: VGPR r holds M = r + 8*half, N = l16 (cols +16 for *1 accs)
#pragma unroll
    for (int r = 0; r < 8; ++r) {
        const int row = row0 + r + half * 8;
        if (row < N) {
            xl[(size_t)row * HDIM + l16]      = cl0[r];
            xl[(size_t)row * HDIM + 16 + l16] = cl1[r];
            xr[(size_t)row * HDIM + l16]      = cr0[r];
            xr[(size_t)row * HDIM + 16 + l16] = cr1[r];
        }
    }
}

// Scalar fallback for unexpected Cin (not used for the reference shapes).
__global__ void gemm_xlxr_scalar(const float* __restrict__ x,
                                 const float* __restrict__ Wl,
                                 const float* __restrict__ Wr,
                                 float* __restrict__ xl, float* __restrict__ xr,
                                 int N, int Cin) {
    int idx = blockIdx.x * blockDim.x + threadIdx.x;
    if (idx >= N * HDIM) return;
    int row = idx >> 5, n = idx & 31;
    float al = 0.f, ar = 0.f;
    for (int k = 0; k < Cin; ++k) {
        float v = x[(size_t)row * Cin + k];
        al += v * Wl[(size_t)k * HDIM + n];
        ar += v * Wr[(size_t)k * HDIM + n];
    }
    xl[idx] = al; xr[idx] = ar;
}

// ---------------------------------------------------------------------------
// Edge pass A: logit_e = att . leakyrelu(xl[src] + xr[dst]); segment max
// Edges e < E come from edge_index; e in [E, E+N) are self loops.
// ---------------------------------------------------------------------------
__global__ void edge_logits_kernel(const int* __restrict__ ei, int E, int N,
                                   const float* __restrict__ xl,
                                   const float* __restrict__ xr,
                                   const float* __restrict__ att,
                                   float* __restrict__ logit,
                                   float* __restrict__ mmax) {
    int e = blockIdx.x * blockDim.x + threadIdx.x;
    if (e >= E + N) return;
    int s, d;
    if (e < E) { s = ei[e]; d = ei[(size_t)E + e]; }
    else       { s = d = e - E; }

    const float4* ps = (const float4*)(xl + (size_t)s * HDIM);
    const float4* pd = (const float4*)(xr + (size_t)d * HDIM);
    const float4* pa = (const float4*)att;
    float acc = 0.f;
#pragma unroll
    for (int q = 0; q < 8; ++q) {
        float4 a = ps[q], b = pd[q], w = pa[q];
        float v;
        v = a.x + b.x; v = (v > 0.f) ? v : v * NEG_SLOPE; acc += v * w.x;
        v = a.y + b.y; v = (v > 0.f) ? v : v * NEG_SLOPE; acc += v * w.y;
        v = a.z + b.z; v = (v > 0.f) ? v : v * NEG_SLOPE; acc += v * w.z;
        v = a.w + b.w; v = (v > 0.f) ? v : v * NEG_SLOPE; acc += v * w.w;
    }
    logit[e] = acc;
    atomicMax(mmax + d, acc);   // GLOBAL_ATOMIC_MAX_NUM_F32
}

// ---------------------------------------------------------------------------
// Edge pass B: w = exp(logit - m[dst]); denom[dst]+=w; num[dst,:]+=w*xl[src,:]
// ---------------------------------------------------------------------------
__global__ void edge_accum_kernel(const int* __restrict__ ei, int E, int N,
                                  const float* __restrict__ xl,
                                  const float* __restrict__ logit,
                                  const float* __restrict__ mmax,
                                  float* __restrict__ denom,
                                  float* __restrict__ num) {
    int e = blockIdx.x * blockDim.x + threadIdx.x;
    if (e >= E + N) return;
    int s, d;
    if (e < E) { s = ei[e]; d = ei[(size_t)E + e]; }
    else       { s = d = e - E; }

    float w = __expf(logit[e] - mmax[d]);
    atomicAdd(denom + d, w);
    const float4* p = (const float4*)(xl + (size_t)s * HDIM);
    float* q = num + (size_t)d * HDIM;
#pragma unroll
    for (int qi = 0; qi < 8; ++qi) {
        float4 v = p[qi];
        atomicAdd(q + 4 * qi + 0, w * v.x);
        atomicAdd(q + 4 * qi + 1, w * v.y);
        atomicAdd(q + 4 * qi + 2, w * v.z);
        atomicAdd(q + 4 * qi + 3, w * v.w);
    }
}

// ---------------------------------------------------------------------------
// Node finalize: out = relu(num/denom + bias)   (denom>0: self loops)
// ---------------------------------------------------------------------------
__global__ void node_finalize_kernel(const float* __restrict__ num,
                                     const float* __restrict__ denom,
                                     const float* __restrict__ bias,
                                     float* __restrict__ out, int N) {
    int idx = blockIdx.x * blockDim.x + threadIdx.x;
    if (idx >= N * HDIM) return;
    int i = idx >> 5, h = idx & 31;
    float v = num[idx] / denom[i] + bias[h];
    out[idx] = (v > 0.f) ? v : 0.f;
}

// ---------------------------------------------------------------------------
// global_mean_pool accumulation + readout
// ---------------------------------------------------------------------------
__global__ void pool_accum_kernel(const float* __restrict__ h2,
                                  const int* __restrict__ batch,
                                  float* __restrict__ gsum,
                                  float* __restrict__ gcnt, int N) {
    int idx = blockIdx.x * blockDim.x + threadIdx.x;
    if (idx >= N * HDIM) return;
    int i = idx >> 5, h = idx & 31;
    int g = batch[i];
    atomicAdd(gsum + (size_t)g * HDIM + h, h2[idx]);
    if (h == 0) atomicAdd(gcnt + g, 1.f);
}

__global__ void pool_final_kernel(const float* __restrict__ gsum,
                                  const float* __restrict__ gcnt,
                                  float* __restrict__ feat_out) {
    int g = blockIdx.x, t = threadIdx.x;   // 512 x 32
    float c = gcnt[g];
    if (c < 1.f) c = 1.f;
    feat_out[(size_t)g * HDIM + t] = gsum[(size_t)g * HDIM + t] / c;
}

__global__ void head_logits_kernel(const float* __restrict__ feat,
                                   const float* __restrict__ Wlin,
                                   const float* __restrict__ blin,
                                   float* __restrict__ logits_out, int Cout) {
    int idx = blockIdx.x * blockDim.x + threadIdx.x;  // NUM_GRAPHS * Cout
    if (idx >= NUM_GRAPHS * Cout) return;
    int g = idx / Cout, c = idx % Cout;
    float acc = blin[c];
#pragma unroll
    for (int h = 0; h < HDIM; ++h)
        acc += feat[(size_t)g * HDIM + h] * Wlin[(size_t)h * Cout + c];
    logits_out[idx] = acc;
}

// ---------------------------------------------------------------------------
static void launch_gemm(const float* x, const float* Wl, const float* Wr,
                        float* xl, float* xr, int N, int Cin, hipStream_t stream) {
    const int gemm_waves  = (N + 15) / 16;
    const int gemm_blocks = (gemm_waves + 7) / 8;   // 8 waves / block
    if (Cin == 19)
        gemm_xlxr_wmma<19><<<gemm_blocks, 256, 0, stream>>>(x, Wl, Wr, xl, xr, N);
    else if (Cin == 32)
        gemm_xlxr_wmma<32><<<gemm_blocks, 256, 0, stream>>>(x, Wl, Wr, xl, xr, N);
    else
        gemm_xlxr_scalar<<<(N * HDIM + 255) / 256, 256, 0, stream>>>(x, Wl, Wr, xl, xr, N, Cin);
}

extern "C" void kernel_launch(void* const* d_in, const int* in_sizes, int n_in,
                              void* d_out, int out_size, void* d_ws, size_t ws_size,
                              hipStream_t stream) {
    const float* x     = (const float*)d_in[0];
    const int*   ei    = (const int*)  d_in[1];
    const int*   batch = (const int*)  d_in[2];
    const float* Wl1   = (const float*)d_in[3];
    const float* Wr1   = (const float*)d_in[4];
    const float* att1  = (const float*)d_in[5];
    const float* b1    = (const float*)d_in[6];
    const float* Wl2   = (const float*)d_in[7];
    const float* Wr2   = (const float*)d_in[8];
    const float* att2  = (const float*)d_in[9];
    const float* b2    = (const float*)d_in[10];
    const float* Wlin  = (const float*)d_in[11];
    const float* blin  = (const float*)d_in[12];

    const int N    = in_sizes[2];            // 100000
    const int Cin  = in_sizes[0] / N;        // 19
    const int E    = in_sizes[1] / 2;        // 3200000
    const int Cout = in_sizes[12];           // 4
    const int Nh   = N * HDIM;

    // Workspace layout (floats)
    float* ws    = (float*)d_ws;
    float* xl    = ws;                 // N*32
    float* xr    = xl    + Nh;         // N*32
    float* num   = xr    + Nh;         // N*32
    float* h1    = num   + Nh;         // N*32
    float* h2    = h1    + Nh;         // N*32
    float* logit = h2    + Nh;         // E+N
    float* mmax  = logit + (E + N);    // N
    float* denom = mmax  + N;          // N
    float* gsum  = denom + N;          // 512*32
    float* gcnt  = gsum  + NUM_GRAPHS * HDIM; // 512

    float* out_logits = (float*)d_out;                     // 512*4
    float* out_feat   = out_logits + NUM_GRAPHS * Cout;    // 512*32

    const int TPB = 256;
    const int edge_blocks = (E + N + TPB - 1) / TPB;
    const int node_blocks = (Nh + TPB - 1) / TPB;

    // ---- Layer 1 ----
    init_kernel<<<node_blocks, TPB, 0, stream>>>(num, denom, mmax, gsum, gcnt, N);
    launch_gemm(x, Wl1, Wr1, xl, xr, N, Cin, stream);
    edge_logits_kernel<<<edge_blocks, TPB, 0, stream>>>(ei, E, N, xl, xr, att1, logit, mmax);
    edge_accum_kernel<<<edge_blocks, TPB, 0, stream>>>(ei, E, N, xl, logit, mmax, denom, num);
    node_finalize_kernel<<<node_blocks, TPB, 0, stream>>>(num, denom, b1, h1, N);

    // ---- Layer 2 ----
    init_kernel<<<node_blocks, TPB, 0, stream>>>(num, denom, mmax, gsum, gcnt, N);
    launch_gemm(h1, Wl2, Wr2, xl, xr, N, HDIM, stream);
    edge_logits_kernel<<<edge_blocks, TPB, 0, stream>>>(ei, E, N, xl, xr, att2, logit, mmax);
    edge_accum_kernel<<<edge_blocks, TPB, 0, stream>>>(ei, E, N, xl, logit, mmax, denom, num);
    node_finalize_kernel<<<node_blocks, TPB, 0, stream>>>(num, denom, b2, h2, N);

    // ---- Readout ----
    pool_accum_kernel<<<node_blocks, TPB, 0, stream>>>(h2, batch, gsum, gcnt, N);
    pool_final_kernel<<<NUM_GRAPHS, HDIM, 0, stream>>>(gsum, gcnt, out_feat);
    head_logits_kernel<<<(NUM_GRAPHS * Cout + TPB - 1) / TPB, TPB, 0, stream>>>(
        out_feat, Wlin, blin, out_logits, Cout);
}